// GNNBlock_1675037245477
// MI455X (gfx1250) — compile-verified
//
#include <hip/hip_runtime.h>
#include <hip/hip_bf16.h>

// ---------------- WMMA types ----------------
typedef float v2f __attribute__((ext_vector_type(2)));
typedef float v8f __attribute__((ext_vector_type(8)));

#define NEG_SLOPE 0.2f

// ---------------- helpers ----------------
__device__ __forceinline__ void atomicMaxFloat(float* addr, float val) {
    // sign-aware monotone mapping: works for all finite floats and -inf init
    if (val >= 0.0f) atomicMax((int*)addr, __float_as_int(val));
    else             atomicMin((unsigned int*)addr, __float_as_uint(val));
}

// ---------------- init kernels ----------------
__global__ void init_out_kernel(float* __restrict__ out, const float* __restrict__ bias,
                                const float* __restrict__ rel, int n32, int relsz) {
    int t = blockIdx.x * blockDim.x + threadIdx.x;
    if (t < n32)               out[t] = bias[t & 31];
    else if (t < n32 + relsz)  out[t] = rel[t - n32];
}

__global__ void fill_kernel(float* __restrict__ p, float v, int count) {
    int t = blockIdx.x * blockDim.x + threadIdx.x;
    if (t < count) p[t] = v;
}

// ---------------- edge_attr mean scatter (for self-loop fill) ----------------
__global__ void scatter_ea_kernel(const int* __restrict__ ei, const int* __restrict__ ridx,
                                  const float* __restrict__ rel,
                                  float* __restrict__ ea_sum, float* __restrict__ cnt, int E) {
    int t = blockIdx.x * blockDim.x + threadIdx.x;
    if (t >= E * 16) return;
    int e = t >> 4, j = t & 15;                 // j indexes float4 chunk of 64-dim row
    int dn = ei[E + e];
    int r  = ridx[e];
    float4 v = ((const float4*)(rel + (size_t)r * 64))[j];
    float* pd = ea_sum + (size_t)dn * 64 + j * 4;
    atomicAdd(pd + 0, v.x); atomicAdd(pd + 1, v.y);
    atomicAdd(pd + 2, v.z); atomicAdd(pd + 3, v.w);
    if (j == 0) atomicAdd(cnt + dn, 1.0f);
}

__global__ void invert_cnt_kernel(float* __restrict__ cnt, int n) {
    int t = blockIdx.x * blockDim.x + threadIdx.x;
    if (t < n) cnt[t] = 1.0f / fmaxf(cnt[t], 1.0f);
}

// ---- fp32 WMMA GEMM: C[M,256] = scale(A)[M,K] @ W[K,256] (+bias), K compile-time ----
// block = 256 threads (8 waves). blockIdx.x = row tile (16 rows).
// Each wave keeps two accumulators: n-tiles {wave, wave+8}; A fragment loaded once
// per k-step (single b64) and fed to both WMMAs. A rows clamped (never stored OOB),
// so no predication inside the loop and EXEC stays all-ones for WMMA.
// HAS_SCALE / HAS_BIAS templated so unused VALU work vanishes from the hot loop.
template <int K, bool HAS_SCALE, bool HAS_BIAS>
__global__ void wmma_gemm256_kernel(const float* __restrict__ A,
                                    const float* __restrict__ rowscale,
                                    const float* __restrict__ W,
                                    const float* __restrict__ bias,
                                    float* __restrict__ C, int M) {
    const int wave = threadIdx.x >> 5;
    const int lane = threadIdx.x & 31;
    const int m0   = blockIdx.x * 16;
    const int lrow = lane & 15;   // m-index for A loads, n-index for B/C
    const int lhi  = lane >> 4;   // half-wave selector: k offset 0 or 2
    const int arow = m0 + lrow;
    const int arow_c = (arow < M) ? arow : (M - 1);   // clamp: garbage rows never stored

    float scale = 1.0f;
    if (HAS_SCALE) scale = rowscale[arow_c];

    const float* __restrict__ Arow = A + (size_t)arow_c * K + 2 * lhi;
    const int n0a = wave * 16;          // first n-tile
    const int n0b = n0a + 128;          // second n-tile
    const float* __restrict__ Wa = W + n0a + lrow + (size_t)2 * lhi * 256;
    const float* __restrict__ Wb = W + n0b + lrow + (size_t)2 * lhi * 256;

    v8f acc0 = {}, acc1 = {};
#pragma unroll
    for (int kk = 0; kk < K; kk += 4) {
        v2f a = *(const v2f*)(Arow + kk);            // k = kk+2*lhi, kk+2*lhi+1
        if (HAS_SCALE) a *= scale;
        v2f b0, b1;
        b0.x = Wa[(size_t)kk * 256];
        b0.y = Wa[(size_t)kk * 256 + 256];
        b1.x = Wb[(size_t)kk * 256];
        b1.y = Wb[(size_t)kk * 256 + 256];
        acc0 = __builtin_amdgcn_wmma_f32_16x16x4_f32(
            false, a, false, b0, (short)0, acc0, false, false);
        acc1 = __builtin_amdgcn_wmma_f32_16x16x4_f32(
            false, a, false, b1, (short)0, acc1, false, false);
    }
    float bva = 0.0f, bvb = 0.0f;
    if (HAS_BIAS) { bva = bias[n0a + lrow]; bvb = bias[n0b + lrow]; }
#pragma unroll
    for (int v = 0; v < 8; ++v) {                    // C: m = v + 8*lhi, n = lrow
        int row = m0 + v + 8 * lhi;
        if (row < M) {
            C[(size_t)row * 256 + n0a + lrow] = HAS_BIAS ? (acc0[v] + bva) : acc0[v];
            C[(size_t)row * 256 + n0b + lrow] = HAS_BIAS ? (acc1[v] + bvb) : acc1[v];
        }
    }
}

// ---------------- per-edge attention logits + segment max ----------------
// one wave per (edge or self-loop). lane covers dims [lane*8, lane*8+8) of 256.
__global__ void edge_logits_kernel(const int* __restrict__ ei, const int* __restrict__ ridx,
                                   const float* __restrict__ xl, const float* __restrict__ xr,
                                   const float* __restrict__ relp, const float* __restrict__ lep,
                                   const float* __restrict__ att,
                                   float* __restrict__ logits, float* __restrict__ lmax,
                                   int E, int Etot) {
    int eid = blockIdx.x * (blockDim.x >> 5) + (threadIdx.x >> 5);
    if (eid >= Etot) return;
    int lane = threadIdx.x & 31;
    int s, dn; const float* ep;
    if (eid < E) { s = ei[eid]; dn = ei[E + eid]; ep = relp + (size_t)ridx[eid] * 256; }
    else         { s = dn = eid - E;              ep = lep  + (size_t)s * 256; }

    const float* pl = xl + (size_t)s  * 256 + lane * 8;
    const float* pr = xr + (size_t)dn * 256 + lane * 8;
    const float* pe = ep + lane * 8;
    const int h = lane >> 2;                         // 4 lanes per head
    const float* pa = att + h * 32 + (lane & 3) * 8;

    float lv[8], rv[8], ev[8], av[8];
    *(float4*)&lv[0] = *(const float4*)(pl);  *(float4*)&lv[4] = *(const float4*)(pl + 4);
    *(float4*)&rv[0] = *(const float4*)(pr);  *(float4*)&rv[4] = *(const float4*)(pr + 4);
    *(float4*)&ev[0] = *(const float4*)(pe);  *(float4*)&ev[4] = *(const float4*)(pe + 4);
    *(float4*)&av[0] = *(const float4*)(pa);  *(float4*)&av[4] = *(const float4*)(pa + 4);

    float partial = 0.0f;
#pragma unroll
    for (int j = 0; j < 8; ++j) {
        float v = lv[j] + rv[j] + ev[j];
        v = (v >= 0.0f) ? v : NEG_SLOPE * v;         // leaky_relu
        partial += v * av[j];
    }
    partial += __shfl_xor(partial, 1, 32);
    partial += __shfl_xor(partial, 2, 32);
    if ((lane & 3) == 0) {
        logits[(size_t)eid * 8 + h] = partial;
        atomicMaxFloat(lmax + (size_t)dn * 8 + h, partial);
    }
}

// ---------------- exp(logit - max) and segment denominator ----------------
__global__ void exp_denom_kernel(const int* __restrict__ ei, float* __restrict__ logits,
                                 const float* __restrict__ lmax, float* __restrict__ denom,
                                 int E, int Etot) {
    int t = blockIdx.x * blockDim.x + threadIdx.x;
    if (t >= Etot * 8) return;
    int eid = t >> 3, h = t & 7;
    int dn = (eid < E) ? ei[E + eid] : (eid - E);
    float ex = __expf(logits[t] - lmax[(size_t)dn * 8 + h]);
    logits[t] = ex;                                   // overwrite with ex
    atomicAdd(denom + (size_t)dn * 8 + h, ex);
}

// ---- weighted aggregation: out[dst,d] += (1/8) sum_h alpha_h * xl[src,h,d] ----
__global__ void aggregate_kernel(const int* __restrict__ ei, const float* __restrict__ xl,
                                 const float* __restrict__ ex, const float* __restrict__ denom,
                                 float* __restrict__ out, int E, int Etot) {
    int eid = blockIdx.x * (blockDim.x >> 5) + (threadIdx.x >> 5);
    if (eid >= Etot) return;
    int d = threadIdx.x & 31;
    int s, dn;
    if (eid < E) { s = ei[eid]; dn = ei[E + eid]; }
    else         { s = dn = eid - E; }
    const float* pex  = ex    + (size_t)eid * 8;
    const float* pden = denom + (size_t)dn  * 8;
    const float* pxl  = xl + (size_t)s * 256 + d;

    float w[8];
#pragma unroll
    for (int h = 0; h < 8; ++h)
        w[h] = pex[h] / (pden[h] + 1e-16f);

    float acc = 0.0f;
#pragma unroll
    for (int h = 0; h < 8; ++h)
        acc += w[h] * pxl[h * 32];
    atomicAdd(out + (size_t)dn * 32 + d, 0.125f * acc);
}

// ---------------- launcher ----------------
static inline int cdiv(long long a, long long b) { return (int)((a + b - 1) / b); }

extern "C" void kernel_launch(void* const* d_in, const int* in_sizes, int n_in,
                              void* d_out, int out_size, void* d_ws, size_t ws_size,
                              hipStream_t stream) {
    const float* x         = (const float*)d_in[0];
    const int*   ei        = (const int*)  d_in[1];   // [2,E] flat: src then dst
    const float* relations = (const float*)d_in[2];   // [NREL,64]
    const int*   ridx      = (const int*)  d_in[3];   // [E]
    const float* W_l       = (const float*)d_in[4];   // [128,256]
    const float* b_l       = (const float*)d_in[5];
    const float* W_r       = (const float*)d_in[6];
    const float* b_r       = (const float*)d_in[7];
    const float* W_e       = (const float*)d_in[8];   // [64,256]
    const float* att       = (const float*)d_in[9];   // [8,32]
    const float* bias      = (const float*)d_in[10];  // [32]

    const int n    = in_sizes[0] / 128;
    const int E    = in_sizes[1] / 2;
    const int nrel = in_sizes[2] / 64;
    const int Etot = E + n;

    float* out = (float*)d_out;
    float* ws  = (float*)d_ws;

    // workspace layout (floats)
    size_t o_xl     = 0;
    size_t o_xr     = o_xl     + (size_t)n * 256;
    size_t o_lep    = o_xr     + (size_t)n * 256;
    size_t o_relp   = o_lep    + (size_t)n * 256;
    size_t o_logits = o_relp   + (size_t)nrel * 256;
    size_t o_ea     = o_logits + (size_t)Etot * 8;
    size_t o_cnt    = o_ea     + (size_t)n * 64;
    size_t o_denom  = o_cnt    + (size_t)n;
    size_t o_lmax   = o_denom  + (size_t)n * 8;

    const int B = 256;

    // 1) init output (bias + passthrough of relations) and workspace
    {
        int tot = n * 32 + nrel * 64;
        init_out_kernel<<<cdiv(tot, B), B, 0, stream>>>(out, bias, relations, n * 32, nrel * 64);
        int zcnt = (int)(n * 64 + n + (size_t)n * 8);   // ea_sum, cnt, denom contiguous
        fill_kernel<<<cdiv(zcnt, B), B, 0, stream>>>(ws + o_ea, 0.0f, zcnt);
        fill_kernel<<<cdiv(n * 8, B), B, 0, stream>>>(ws + o_lmax, -INFINITY, n * 8);
    }

    // 2) scatter relation rows per destination (self-loop 'mean' fill)
    scatter_ea_kernel<<<cdiv((long long)E * 16, B), B, 0, stream>>>(
        ei, ridx, relations, ws + o_ea, ws + o_cnt, E);
    invert_cnt_kernel<<<cdiv(n, B), B, 0, stream>>>(ws + o_cnt, n);

    // 3) WMMA GEMMs (fp32 16x16x4, K fully unrolled)
    wmma_gemm256_kernel<128, false, true><<<cdiv(n, 16), B, 0, stream>>>(
        x, nullptr, W_l, b_l, ws + o_xl, n);
    wmma_gemm256_kernel<128, false, true><<<cdiv(n, 16), B, 0, stream>>>(
        x, nullptr, W_r, b_r, ws + o_xr, n);
    wmma_gemm256_kernel<64, false, false><<<cdiv(nrel, 16), B, 0, stream>>>(
        relations, nullptr, W_e, nullptr, ws + o_relp, nrel);
    wmma_gemm256_kernel<64, true, false><<<cdiv(n, 16), B, 0, stream>>>(
        ws + o_ea, ws + o_cnt, W_e, nullptr, ws + o_lep, n);

    // 4) attention logits + segment max (one wave per edge)
    edge_logits_kernel<<<cdiv(Etot, B / 32), B, 0, stream>>>(
        ei, ridx, ws + o_xl, ws + o_xr, ws + o_relp, ws + o_lep, att,
        ws + o_logits, ws + o_lmax, E, Etot);

    // 5) exp + segment sum
    exp_denom_kernel<<<cdiv((long long)Etot * 8, B), B, 0, stream>>>(
        ei, ws + o_logits, ws + o_lmax, ws + o_denom, E, Etot);

    // 6) softmax-weighted aggregation + head mean (one wave per edge)
    aggregate_kernel<<<cdiv(Etot, B / 32), B, 0, stream>>>(
        ei, ws + o_xl, ws + o_logits, ws + o_denom, out, E, Etot);
}